// TAU_20057497272937
// MI455X (gfx1250) — compile-verified
//
#include <hip/hip_runtime.h>
#include <math.h>

// ---------------------------------------------------------------------------
// Problem constants (from reference): N=131072, d=8, m=1024, E=64, sigma=1
// d_out = [ ensemble (N floats) | net_out (N floats) ]
// ---------------------------------------------------------------------------

typedef float v2f __attribute__((ext_vector_type(2)));
typedef float v8f __attribute__((ext_vector_type(8)));

#define DIM       8
#define M_CENTERS 1024
#define ENS       64

// ---------------------------------------------------------------------------
// Kernel 1: ensemble[row] = sum_e probs[row][e] * w[e]   (HBM-bound matvec)
// One wave per row: lane l handles columns l and l+32 -> coalesced 128B
// requests, shfl-xor tree reduction.
// ---------------------------------------------------------------------------
__global__ __launch_bounds__(256) void ensemble_kernel(
    const float* __restrict__ probs,
    const float* __restrict__ w,
    float* __restrict__ out) {
  const int lane = threadIdx.x & 31;
  const int row  = blockIdx.x * (blockDim.x >> 5) + (threadIdx.x >> 5);

  const float* pr = probs + (size_t)row * ENS;
  float s = pr[lane] * w[lane] + pr[lane + 32] * w[lane + 32];
#pragma unroll
  for (int off = 16; off >= 1; off >>= 1)
    s += __shfl_xor(s, off, 32);
  if (lane == 0) out[row] = s;
}

// ---------------------------------------------------------------------------
// Kernel 2: net_out[row] = sum_n exp2( (x_r . c_n)*log2e + p_r + q_n )
// with p_r = -0.5*||x_r||^2*log2e folded into the WMMA C operand and the
// log2e scale folded into the A fragment, so the matrix pipe computes the
// entire exp2 argument. Inner loop per element: v_exp_f32 + v_add_f32.
//
// Fragment layouts (wave32, ISA 7.12.2):
//   A 16x4 f32 : lane = 16*h + n holds M=n, VGPR0=K(2h), VGPR1=K(2h+1)
//   B 4x16 f32 : lane = 16*h + n holds N=n, VGPR0=K(2h), VGPR1=K(2h+1)
//   C/D 16x16  : VGPR i -> M = i + 8*h, N = n
// ---------------------------------------------------------------------------
__global__ __launch_bounds__(256) void netout_kernel(
    const float* __restrict__ x,
    const float* __restrict__ centers,
    const float* __restrict__ coeff,
    float* __restrict__ net_out) {
  __shared__ float red[8];
  __shared__ float qhat[M_CENTERS];

  const int tid  = threadIdx.x;
  const int lane = tid & 31;
  const int wv   = tid >> 5;
  const int n    = lane & 15;       // tile column (and A-row index)
  const int h    = lane >> 4;       // K-half / M-half served by this lane
  const int row0 = blockIdx.x * 128 + wv * 16;

  const float LOG2E = 1.4426950408889634f;
  // log2(norm_const) = log2((2*pi)^-4) = -4*log2(2*pi)
  const float LOG2_NORM = -10.605984517889275f;

  // ---- block-wide softmax statistics over the 1024 coefficients ----
  float tmax = -3.402823466e38f;
#pragma unroll
  for (int j = 0; j < M_CENTERS / 256; ++j)
    tmax = fmaxf(tmax, coeff[tid + 256 * j]);
#pragma unroll
  for (int off = 16; off >= 1; off >>= 1)
    tmax = fmaxf(tmax, __shfl_xor(tmax, off, 32));
  if (lane == 0) red[wv] = tmax;
  __syncthreads();
  float cmax = red[0];
#pragma unroll
  for (int j = 1; j < 8; ++j) cmax = fmaxf(cmax, red[j]);
  __syncthreads();

  float tsum = 0.0f;
#pragma unroll
  for (int j = 0; j < M_CENTERS / 256; ++j)
    tsum += __expf(coeff[tid + 256 * j] - cmax);
#pragma unroll
  for (int off = 16; off >= 1; off >>= 1)
    tsum += __shfl_xor(tsum, off, 32);
  if (lane == 0) red[wv] = tsum;
  __syncthreads();
  float csum = red[0];
#pragma unroll
  for (int j = 1; j < 8; ++j) csum += red[j];

  const float qbias = LOG2_NORM - __log2f(csum);

  // ---- q table: q_n = (coeff_n - cmax - 0.5*||c_n||^2)*log2e + qbias ----
#pragma unroll
  for (int j = 0; j < M_CENTERS / 256; ++j) {
    const int c = tid + 256 * j;
    const float* cr = centers + (size_t)c * DIM;
    float c2 = 0.0f;
#pragma unroll
    for (int k = 0; k < DIM; ++k) c2 += cr[k] * cr[k];
    qhat[c] = (coeff[c] - cmax - 0.5f * c2) * LOG2E + qbias;
  }
  __syncthreads();

  // ---- p_r = -0.5*||x_r||^2*log2e for the 16 rows of this wave's tile ----
  float x2full = 0.0f;
  {
    const float* xr = x + (size_t)(row0 + n) * DIM;
#pragma unroll
    for (int k = 0; k < DIM; ++k) x2full += xr[k] * xr[k];
  }
  float phat[8];
#pragma unroll
  for (int i = 0; i < 8; ++i)
    phat[i] = -0.5f * LOG2E * __shfl(x2full, i + 8 * h, 32);  // row M = i+8h

  // ---- A fragments (x rows), pre-scaled by log2e ----
  const float2* xr2 = (const float2*)(x + (size_t)(row0 + n) * DIM);
  v2f a_lo, a_hi;
  {
    float2 t0 = xr2[h];       // K = 2h, 2h+1
    float2 t1 = xr2[2 + h];   // K = 4+2h, 5+2h
    a_lo.x = t0.x * LOG2E; a_lo.y = t0.y * LOG2E;
    a_hi.x = t1.x * LOG2E; a_hi.y = t1.y * LOG2E;
  }

  float acc[8];
#pragma unroll
  for (int i = 0; i < 8; ++i) acc[i] = 0.0f;

  // ---- sweep all 1024 centers in 16-wide tiles ----
  for (int ct = 0; ct < M_CENTERS / 16; ++ct) {
    const int c0 = ct * 16;
    const float2* cr2 = (const float2*)(centers + (size_t)(c0 + n) * DIM);
    float2 t0 = cr2[h];       // K = 2h, 2h+1 of center column n
    float2 t1 = cr2[2 + h];   // K = 4+2h, 5+2h
    v2f b_lo, b_hi;
    b_lo.x = t0.x; b_lo.y = t0.y;
    b_hi.x = t1.x; b_hi.y = t1.y;

    const float qn = qhat[c0 + n];

    // Seed C with the full exp2 bias: D = A.B*log2e + (p_r + q_n)
    v8f dmat;
#pragma unroll
    for (int i = 0; i < 8; ++i) dmat[i] = phat[i] + qn;

    dmat = __builtin_amdgcn_wmma_f32_16x16x4_f32(
        false, a_lo, false, b_lo, (short)0, dmat, false, false);
    dmat = __builtin_amdgcn_wmma_f32_16x16x4_f32(
        false, a_hi, false, b_hi, (short)0, dmat, false, false);

    // per element: one v_exp_f32 + one v_add_f32
#pragma unroll
    for (int i = 0; i < 8; ++i)
      acc[i] += __builtin_amdgcn_exp2f(dmat[i]);
  }

  // ---- reduce over the 16 columns held within each lane-half ----
#pragma unroll
  for (int i = 0; i < 8; ++i) {
#pragma unroll
    for (int off = 8; off >= 1; off >>= 1)
      acc[i] += __shfl_xor(acc[i], off, 32);
  }
  if (n == 0) {
    // lane 0 (h=0): rows row0+0..7 ; lane 16 (h=1): rows row0+8..15
#pragma unroll
    for (int i = 0; i < 8; ++i)
      net_out[row0 + 8 * h + i] = acc[i];
  }
}

// ---------------------------------------------------------------------------
// Entry point
// ---------------------------------------------------------------------------
extern "C" void kernel_launch(void* const* d_in, const int* in_sizes, int n_in,
                              void* d_out, int out_size, void* d_ws, size_t ws_size,
                              hipStream_t stream) {
  const float* x       = (const float*)d_in[0];  // (N, 8)
  const float* probs   = (const float*)d_in[1];  // (N, 64)
  const float* weights = (const float*)d_in[2];  // (64,)
  const float* centers = (const float*)d_in[3];  // (1024, 8)
  const float* coeff   = (const float*)d_in[4];  // (1024,)
  float* out = (float*)d_out;

  const int N = in_sizes[0] / DIM;  // 131072

  // ensemble: one wave per row, 8 waves per block
  ensemble_kernel<<<N / 8, 256, 0, stream>>>(probs, weights, out);

  // net_out: 8 waves per block, 16 rows per wave -> 128 rows per block
  netout_kernel<<<N / 128, 256, 0, stream>>>(x, centers, coeff, out + N);
}